// ContrastiveLoss_51797305589975
// MI455X (gfx1250) — compile-verified
//
#include <hip/hip_runtime.h>
#include <hip/hip_bf16.h>

// SupCon loss, fused: z=normalize(feature); sim=z z^T computed as 16x16 WMMA
// tiles (V_WMMA_F32_16X16X4_F32, f32 precision), masks/exp/reductions applied
// on-the-fly. sim (256MB) is never materialized.
// B panel double-buffered in LDS via GLOBAL_LOAD_ASYNC_TO_LDS_B128 (ASYNCcnt)
// so the L2->LDS copy of tile jt+1 overlaps the 32 WMMAs of tile jt.
//
// B=8192, D=128, TEMPERATURE=0.5 (=> exp(2*sim)), EPS=1e-8.

#define BN 8192
#define DD 128
#define KSTEPS 32            // D/4
#define NCHUNK 8             // column chunks
#define JTILES 64            // 512 col-tiles / NCHUNK

typedef __attribute__((ext_vector_type(2))) float v2f;
typedef __attribute__((ext_vector_type(8))) float v8f;
typedef __attribute__((ext_vector_type(4))) int   v4i;

#if __has_builtin(__builtin_amdgcn_global_load_async_to_lds_b128) && \
    __has_builtin(__builtin_amdgcn_s_wait_asynccnt)
#define HAVE_ASYNC_LDS 1
#else
#define HAVE_ASYNC_LDS 0
#endif

#define AS1 __attribute__((address_space(1)))
#define AS3 __attribute__((address_space(3)))

// ---------------- Kernel 1: normalize rows + pack into WMMA f32 16x4 lane layout.
// Packed layout (float2 units): zw2[((blk16*32 + t)*32 + lane)]
//   lane<16 : { z[blk16*16+lane][4t+0], z[..][4t+1] }
//   lane>=16: { z[blk16*16+lane-16][4t+2], z[..][4t+3] }
// This is simultaneously the A-matrix and B-matrix lane layout for
// V_WMMA_F32_16X16X4_F32 (since B = Z^T of the same matrix).
__global__ void __launch_bounds__(256) scl_normalize_pack(const float* __restrict__ f,
                                                          float* __restrict__ zw) {
    int wave = threadIdx.x >> 5;
    int lane = threadIdx.x & 31;
    int row  = blockIdx.x * 8 + wave;     // one wave per row, 32 lanes * 4 elems
    const float4* frow = (const float4*)(f + (size_t)row * DD);
    float4 v = frow[lane];
    float ss = v.x * v.x + v.y * v.y + v.z * v.z + v.w * v.w;
    #pragma unroll
    for (int off = 16; off > 0; off >>= 1) ss += __shfl_xor(ss, off, 32);
    float inv = 1.0f / fmaxf(sqrtf(ss), 1e-12f);
    v.x *= inv; v.y *= inv; v.z *= inv; v.w *= inv;
    int m  = row >> 4;
    int rl = row & 15;
    float2* zw2 = (float2*)zw;
    // column group t == lane: elems 4*lane .. 4*lane+3
    zw2[((size_t)m * 32 + lane) * 32 + rl]      = make_float2(v.x, v.y);
    zw2[((size_t)m * 32 + lane) * 32 + 16 + rl] = make_float2(v.z, v.w);
}

// ---------------- Kernel 2: fused GEMM + exp + masked reductions.
// grid = (64 rowgroups, NCHUNK chunks), block = 256 (8 waves).
// Wave w of rowgroup g owns row-block mblk = g*8+w (16 rows, A strip in regs).
// Block stages each B panel (16 cols x 128 K = 8KB) in LDS, shared by 8 waves.
__global__ void __launch_bounds__(256) scl_main(const float* __restrict__ zw,
                                                const int* __restrict__ label,
                                                float* __restrict__ rowNumP,
                                                float* __restrict__ rowDenP,
                                                float* __restrict__ posP,
                                                float* __restrict__ negP) {
    __shared__ float2 Bs[2][KSTEPS * 32];  // 2 x 8 KB panels (double buffer)
    __shared__ float  wsum[16];

    int wave = threadIdx.x >> 5;
    int lane = threadIdx.x & 31;
    int mblk  = blockIdx.x * 8 + wave;     // 0..511
    int chunk = blockIdx.y;                // 0..7
    const float2* zw2 = (const float2*)zw;

    // A strip: 16 rows x 128 K, WMMA lane layout, 32 float2 = 64 VGPRs
    float2 A[KSTEPS];
    #pragma unroll
    for (int t = 0; t < KSTEPS; ++t)
        A[t] = zw2[((size_t)mblk * 32 + t) * 32 + lane];

    int rowbase = mblk * 16 + ((lane >> 4) << 3);   // C/D: vgpr v -> row rowbase+v
    int labr[8];
    #pragma unroll
    for (int v = 0; v < 8; ++v) labr[v] = label[rowbase + v];

    float num_p[8], den_p[8];
    #pragma unroll
    for (int v = 0; v < 8; ++v) { num_p[v] = 0.0f; den_p[v] = 0.0f; }
    float pos_s = 0.0f, neg_s = 0.0f;

#if HAVE_ASYNC_LDS
    // Prologue: async-copy panel 0 into Bs[0]. Each thread moves 32 bytes of
    // the 8KB panel as two b128 async ops (ASYNCcnt += 2).
    {
        const char* g0 = (const char*)(zw2 + (size_t)(chunk * JTILES) * 1024)
                         + threadIdx.x * 32;
        char* l0 = (char*)&Bs[0][0] + threadIdx.x * 32;
        __builtin_amdgcn_global_load_async_to_lds_b128(
            (AS1 v4i*)g0, (AS3 v4i*)l0, 0, 0);
        __builtin_amdgcn_global_load_async_to_lds_b128(
            (AS1 v4i*)g0, (AS3 v4i*)l0, 16, 0);
    }
#endif

    for (int jt = 0; jt < JTILES; ++jt) {
        int nblk = chunk * JTILES + jt;

#if HAVE_ASYNC_LDS
        __syncthreads();   // all waves done computing jt-1: safe to overwrite Bs[(jt+1)&1]
        if (jt + 1 < JTILES) {
            const char* g = (const char*)(zw2 + (size_t)(nblk + 1) * 1024)
                            + threadIdx.x * 32;
            char* l = (char*)&Bs[(jt + 1) & 1][0] + threadIdx.x * 32;
            __builtin_amdgcn_global_load_async_to_lds_b128(
                (AS1 v4i*)g, (AS3 v4i*)l, 0, 0);
            __builtin_amdgcn_global_load_async_to_lds_b128(
                (AS1 v4i*)g, (AS3 v4i*)l, 16, 0);
            __builtin_amdgcn_s_wait_asynccnt(2);   // panel jt done; jt+1 in flight
        } else {
            __builtin_amdgcn_s_wait_asynccnt(0);
        }
        __syncthreads();   // panel jt visible to all 8 waves
        const float2* Bp = &Bs[jt & 1][0];
#else
        __syncthreads();
        // cooperative panel load: 1024 float2, 4 per thread, coalesced
        #pragma unroll
        for (int k = 0; k < 4; ++k)
            Bs[0][threadIdx.x + k * 256] = zw2[(size_t)nblk * 1024 + threadIdx.x + k * 256];
        __syncthreads();
        const float2* Bp = &Bs[0][0];
#endif

        v8f acc0 = {}; v8f acc1 = {};        // two accumulators to pipeline WMMA RAW chain
        #pragma unroll
        for (int t = 0; t < KSTEPS; t += 2) {
            float2 a0 = A[t], a1 = A[t + 1];
            float2 b0 = Bp[t * 32 + lane];
            float2 b1 = Bp[(t + 1) * 32 + lane];
            v2f av0 = {a0.x, a0.y}, bv0 = {b0.x, b0.y};
            v2f av1 = {a1.x, a1.y}, bv1 = {b1.x, b1.y};
            acc0 = __builtin_amdgcn_wmma_f32_16x16x4_f32(false, av0, false, bv0,
                                                         (short)0, acc0, false, false);
            acc1 = __builtin_amdgcn_wmma_f32_16x16x4_f32(false, av1, false, bv1,
                                                         (short)0, acc1, false, false);
        }

        int col = nblk * 16 + (lane & 15);
        int lc  = label[col];
        #pragma unroll
        for (int v = 0; v < 8; ++v) {
            float s = acc0[v] + acc1[v];          // sim in [-1,1]
            float e = __expf(s * 2.0f);           // exp(sim / 0.5)
            bool same = (labr[v] == lc);
            bool pos  = same && ((rowbase + v) != col);
            num_p[v] += pos  ? e : 0.0f;
            den_p[v] += same ? 0.0f : e;
            pos_s    += pos  ? s : 0.0f;
            neg_s    += same ? 0.0f : s;
        }
    }

    // reduce num/den over the 16 lanes of each half (rows 0-7 low, 8-15 high)
    #pragma unroll
    for (int v = 0; v < 8; ++v) {
        #pragma unroll
        for (int off = 1; off < 16; off <<= 1) {
            num_p[v] += __shfl_xor(num_p[v], off, 32);
            den_p[v] += __shfl_xor(den_p[v], off, 32);
        }
    }
    if ((lane & 15) == 0) {
        #pragma unroll
        for (int v = 0; v < 8; ++v) {
            rowNumP[chunk * BN + rowbase + v] = num_p[v];
            rowDenP[chunk * BN + rowbase + v] = den_p[v];
        }
    }

    // pos/neg: full-wave reduce, then block reduce via LDS (deterministic slots)
    #pragma unroll
    for (int off = 1; off < 32; off <<= 1) {
        pos_s += __shfl_xor(pos_s, off, 32);
        neg_s += __shfl_xor(neg_s, off, 32);
    }
    if (lane == 0) { wsum[wave] = pos_s; wsum[8 + wave] = neg_s; }
    __syncthreads();
    if (threadIdx.x == 0) {
        float p = 0.0f, n = 0.0f;
        #pragma unroll
        for (int w = 0; w < 8; ++w) { p += wsum[w]; n += wsum[8 + w]; }
        int bid = blockIdx.x * NCHUNK + blockIdx.y;   // 0..511
        posP[bid] = p; negP[bid] = n;
    }
}

// ---------------- Kernel 3: finalize (single block, fixed-order => deterministic)
__global__ void __launch_bounds__(256) scl_finalize(const float* __restrict__ rowNumP,
                                                    const float* __restrict__ rowDenP,
                                                    const float* __restrict__ posP,
                                                    const float* __restrict__ negP,
                                                    float* __restrict__ out) {
    __shared__ float red[256];
    float ls = 0.0f;
    for (int r = threadIdx.x; r < BN; r += 256) {
        float num = 0.0f, den = 0.0f;
        #pragma unroll
        for (int c = 0; c < NCHUNK; ++c) {
            num += rowNumP[c * BN + r];
            den += rowDenP[c * BN + r];
        }
        ls += -logf((num + 1e-8f) / (den + num));
    }
    red[threadIdx.x] = ls;
    __syncthreads();
    for (int s = 128; s > 0; s >>= 1) {
        if (threadIdx.x < s) red[threadIdx.x] += red[threadIdx.x + s];
        __syncthreads();
    }
    if (threadIdx.x == 0) {
        float p = 0.0f, n = 0.0f;
        for (int i = 0; i < 512; ++i) { p += posP[i]; n += negP[i]; }
        out[0] = red[0] / (float)BN;
        out[1] = p / 67108864.0f;   // B*B = 8192^2
        out[2] = n / 67108864.0f;
    }
}

extern "C" void kernel_launch(void* const* d_in, const int* in_sizes, int n_in,
                              void* d_out, int out_size, void* d_ws, size_t ws_size,
                              hipStream_t stream) {
    (void)in_sizes; (void)n_in; (void)out_size; (void)ws_size;
    const float* feature = (const float*)d_in[0];
    const int*   label   = (const int*)d_in[1];
    float* out = (float*)d_out;

    char*  ws      = (char*)d_ws;
    float* zw      = (float*)ws;                         // 8192*128 f32 = 4 MiB
    float* rowNumP = (float*)(ws + (4u << 20));          // 8*8192 f32
    float* rowDenP = rowNumP + NCHUNK * BN;              // 8*8192 f32
    float* posP    = rowDenP + NCHUNK * BN;              // 512 f32
    float* negP    = posP + 512;                         // 512 f32

    scl_normalize_pack<<<BN / 8, 256, 0, stream>>>(feature, zw);
    dim3 grid(64, NCHUNK);
    scl_main<<<grid, 256, 0, stream>>>(zw, label, rowNumP, rowDenP, posP, negP);
    scl_finalize<<<1, 256, 0, stream>>>(rowNumP, rowDenP, posP, negP, out);
}